// SimpleMMOEHead_18622978195711
// MI455X (gfx1250) — compile-verified
//
#include <hip/hip_runtime.h>
#include <hip/hip_bf16.h>
#include <math.h>

// ---------------------------------------------------------------------------
// SimpleMMOEHead on MI455X (gfx1250):
//   t = (x @ W_K) @ W_Q            -> two 64x768x768 f32 GEMMs via V_WMMA_F32_16X16X4_F32
//   scores[b,m] = t[b,:] . mem[b,:,m]  -> bandwidth-bound single-pass stream of 402 MB
//   leaky_relu -> atomic scatter by label -> row softmax
// ---------------------------------------------------------------------------

typedef float v2f __attribute__((ext_vector_type(2)));
typedef float v8f __attribute__((ext_vector_type(8)));

#define B_DIM 64
#define D_DIM 768
#define M_DIM 2048
#define C_DIM 1000
#define NEG_SLOPE 0.01f

// One wave32 computes one 16x16 f32 output tile; K-loop steps by 4.
// Row-major A[Md x Kd] * B[Kd x Nd] -> C[Md x Nd].
__global__ __launch_bounds__(128) void gemm16x16x4_f32(
    const float* __restrict__ A, const float* __restrict__ Bm,
    float* __restrict__ C, int Md, int Nd, int Kd) {
  const int wave = threadIdx.x >> 5;
  const int tile = blockIdx.x * (blockDim.x >> 5) + wave;
  const int ntn  = Nd >> 4;
  const int tm   = (tile / ntn) << 4;
  const int tn   = (tile % ntn) << 4;
  if (tm >= Md) return;                       // wave-uniform guard
  const int lane = threadIdx.x & 31;
  const int lo   = lane & 15;                 // M (A) / N (B,C) index
  const int hi   = (lane >> 4) & 1;           // K half select

  v8f acc = {};
  const float* arow = A  + (size_t)(tm + lo) * Kd + 2 * hi;
  const float* bcol = Bm + (size_t)(2 * hi) * Nd + tn + lo;
  #pragma unroll 4
  for (int k = 0; k < Kd; k += 4) {
    v2f a, b;
    // A-frag: A[tm+lo, k + 2*hi + j]   (ISA 16x4 f32 layout)
    a.x = arow[k + 0];
    a.y = arow[k + 1];
    // B-frag: B[k + 2*hi + j, tn+lo]   (ISA 4x16 f32 layout)
    b.x = bcol[(size_t)(k + 0) * Nd];
    b.y = bcol[(size_t)(k + 1) * Nd];
    acc = __builtin_amdgcn_wmma_f32_16x16x4_f32(false, a, false, b,
                                                (short)0, acc, false, false);
  }
  // D layout: acc[v] -> C[tm + v + 8*hi, tn + lo]
  #pragma unroll
  for (int v = 0; v < 8; ++v)
    C[(size_t)(tm + v + 8 * hi) * Nd + tn + lo] = acc[v];
}

__global__ void zero_f32(float* __restrict__ p, int n) {
  int i = blockIdx.x * blockDim.x + threadIdx.x;
  if (i < n) p[i] = 0.0f;
}

// One block per (m-slab, batch). Each thread owns 4 consecutive m's (float4
// loads; lane-adjacent float4s are contiguous in memory => perfect coalescing).
// Streams mem[b,:,:] once, accumulates dot with t[b,:] held in LDS, applies
// leaky-relu, scatters into logits with f32 global atomics.
__global__ __launch_bounds__(128) void scores_scatter(
    const float* __restrict__ T,        // [B, D]
    const float* __restrict__ mem,      // [B, D, M]
    const int*   __restrict__ labels,   // [B, M]
    float*       __restrict__ logits) { // [B, C]
  __shared__ float tsh[D_DIM];
  const int b = blockIdx.y;
  for (int i = threadIdx.x; i < D_DIM; i += blockDim.x)
    tsh[i] = T[b * D_DIM + i];
  __syncthreads();

  const int m0 = (blockIdx.x * blockDim.x + threadIdx.x) << 2;
  const float4* mp =
      (const float4*)(mem + (size_t)b * D_DIM * M_DIM) + (m0 >> 2);
  float4 acc = make_float4(0.f, 0.f, 0.f, 0.f);
  #pragma unroll 8
  for (int e = 0; e < D_DIM; ++e) {
    float4 v = mp[(size_t)e * (M_DIM / 4)];
    if (e + 16 < D_DIM)  // rows are 8KB apart: prefetch ahead (global_prefetch_b8)
      __builtin_prefetch((const void*)&mp[(size_t)(e + 16) * (M_DIM / 4)], 0, 0);
    const float te = tsh[e];
    acc.x = fmaf(te, v.x, acc.x);
    acc.y = fmaf(te, v.y, acc.y);
    acc.z = fmaf(te, v.z, acc.z);
    acc.w = fmaf(te, v.w, acc.w);
  }

  float s[4] = {acc.x, acc.y, acc.z, acc.w};
  #pragma unroll
  for (int j = 0; j < 4; ++j) {
    float v = s[j];
    v = (v >= 0.f) ? v : NEG_SLOPE * v;
    const int lab = labels[(size_t)b * M_DIM + m0 + j];
    atomicAdd(&logits[(size_t)b * C_DIM + lab], v);
  }
}

// One block (8 waves) per batch row: max-reduce, exp-sum, normalize.
__global__ __launch_bounds__(256) void softmax_rows(
    const float* __restrict__ logits, float* __restrict__ out) {
  __shared__ float red[256];
  const int b = blockIdx.x;
  const float* row = logits + (size_t)b * C_DIM;

  float mx = -INFINITY;
  for (int i = threadIdx.x; i < C_DIM; i += 256) mx = fmaxf(mx, row[i]);
  red[threadIdx.x] = mx;
  __syncthreads();
  for (int s = 128; s > 0; s >>= 1) {
    if (threadIdx.x < s)
      red[threadIdx.x] = fmaxf(red[threadIdx.x], red[threadIdx.x + s]);
    __syncthreads();
  }
  mx = red[0];
  __syncthreads();

  float sum = 0.f;
  for (int i = threadIdx.x; i < C_DIM; i += 256) sum += expf(row[i] - mx);
  red[threadIdx.x] = sum;
  __syncthreads();
  for (int s = 128; s > 0; s >>= 1) {
    if (threadIdx.x < s) red[threadIdx.x] += red[threadIdx.x + s];
    __syncthreads();
  }
  const float inv = 1.0f / red[0];

  for (int i = threadIdx.x; i < C_DIM; i += 256)
    out[(size_t)b * C_DIM + i] = expf(row[i] - mx) * inv;
}

extern "C" void kernel_launch(void* const* d_in, const int* in_sizes, int n_in,
                              void* d_out, int out_size, void* d_ws, size_t ws_size,
                              hipStream_t stream) {
  const float* X      = (const float*)d_in[0];  // [64, 768]
  const float* mem    = (const float*)d_in[1];  // [64, 768, 2048]
  const int*   labels = (const int*)d_in[2];    // [64, 2048]
  const float* W_Q    = (const float*)d_in[3];  // [768, 768]
  const float* W_K    = (const float*)d_in[4];  // [768, 768]
  float* out = (float*)d_out;                   // [64, 1000]

  float* Q      = (float*)d_ws;                 // 64*768
  float* T      = Q + B_DIM * D_DIM;            // 64*768
  float* logits = T + B_DIM * D_DIM;            // 64*1000

  // 192 output tiles per GEMM, 4 waves per 128-thread block.
  const int tiles = (B_DIM / 16) * (D_DIM / 16);
  dim3 g1(tiles / 4);
  // Q = X @ W_K
  gemm16x16x4_f32<<<g1, 128, 0, stream>>>(X, W_K, Q, B_DIM, D_DIM, D_DIM);
  // T = Q @ W_Q
  gemm16x16x4_f32<<<g1, 128, 0, stream>>>(Q, W_Q, T, B_DIM, D_DIM, D_DIM);

  zero_f32<<<(B_DIM * C_DIM + 255) / 256, 256, 0, stream>>>(logits,
                                                            B_DIM * C_DIM);

  // Bandwidth-bound main pass: 402 MB streamed once.
  dim3 g2(M_DIM / (128 * 4), B_DIM);  // (4, 64) blocks of 128 threads
  scores_scatter<<<g2, 128, 0, stream>>>(T, mem, labels, logits);

  softmax_rows<<<B_DIM, 256, 0, stream>>>(logits, out);
}